// Auto_Attn_68693706932849
// MI455X (gfx1250) — compile-verified
//
#include <hip/hip_runtime.h>

#define BATCH 4
#define CCH   256
#define NPIX  4096
#define DQ    64

typedef __attribute__((ext_vector_type(2))) float v2f;
typedef __attribute__((ext_vector_type(8))) float v8f;

__device__ __forceinline__ v8f wmma_f32(v2f a, v2f b, v8f c) {
  // D = A(16x4) * B(4x16) + C(16x16), exact fp32
  return __builtin_amdgcn_wmma_f32_16x16x4_f32(false, a, false, b, (short)0, c, false, false);
}

// ---------------------------------------------------------------------------
// Kernel 1: q[b,d,n] = sum_c x[b,c,n] * wq[c,d] + bq[d]
// block: 256 threads handles (b, 16 n). x column tile cached in LDS.
// ---------------------------------------------------------------------------
__global__ void qproj_kernel(const float* __restrict__ x, const float* __restrict__ wq,
                             const float* __restrict__ bq, float* __restrict__ q) {
  __shared__ float xs[CCH][17];
  const int b  = blockIdx.y;
  const int nb = blockIdx.x * 16;
  const int t  = threadIdx.x;
  const int nq = t & 15, cg = t >> 4;
  const float* xb = x + (size_t)b * CCH * NPIX;
#pragma unroll
  for (int i = 0; i < 16; ++i) {
    int c = cg + i * 16;
    xs[c][nq] = xb[(size_t)c * NPIX + nb + nq];  // 16 consecutive n per 16 lanes
  }
  __syncthreads();
#pragma unroll
  for (int jj = 0; jj < 4; ++jj) {
    int d = cg + jj * 16;
    float acc = bq[d];
#pragma unroll 8
    for (int c = 0; c < CCH; ++c)
      acc = fmaf(xs[c][nq], wq[c * DQ + d], acc);
    q[((size_t)b * DQ + d) * NPIX + nb + nq] = acc;  // coalesced over nq
  }
}

// ---------------------------------------------------------------------------
// Kernel 2: per-row softmax stats of energy = q^T q  (online max/sum, WMMA)
// block: 256 threads = 8 waves. wave w: query-subtile (w&3), m-half (w>>2).
// ---------------------------------------------------------------------------
__global__ void stats_kernel(const float* __restrict__ q, float* __restrict__ rmax,
                             float* __restrict__ rinv) {
  __shared__ float qs[2][32][68];     // [m-half][m][d], pitch 68 -> conflict-free b64 reads
  __shared__ float pmx[2][64], psm[2][64];
  const int b   = blockIdx.y;
  const int nbT = blockIdx.x * 64;
  const int t   = threadIdx.x;
  const int w = t >> 5, l = t & 31;
  const int nsub = w & 3, mh = w >> 2;
  const int half = l >> 4, li = l & 15;
  const float* qb = q + (size_t)b * DQ * NPIX;

  // A operand: this wave's 16 query rows, all 64 d (held in regs for whole kernel)
  v2f a[16];
  {
    int nrow = nbT + nsub * 16 + li;
#pragma unroll
    for (int kk = 0; kk < 16; ++kk) {
      int d0 = kk * 4 + 2 * half;
      a[kk].x = qb[(size_t)d0 * NPIX + nrow];
      a[kk].y = qb[(size_t)(d0 + 1) * NPIX + nrow];
    }
  }
  float mx[8], sm[8];
#pragma unroll
  for (int v = 0; v < 8; ++v) { mx[v] = -3.0e38f; sm[v] = 0.0f; }

  const int myh  = t >> 7;        // which m-half this thread stages
  const int tt   = t & 127;
  const int dq8  = tt & 7;        // m-quad 0..7
  const int dr   = tt >> 3;       // d-row group 0..15

  for (int it = 0; it < 64; ++it) {
    __syncthreads();
    // stage 32-wide Q m-tile (transposed to [m][d]) for each m-half
    int mb = myh * 2048 + it * 32;
#pragma unroll
    for (int i = 0; i < 4; ++i) {
      int d = dr + 16 * i;
      float4 vv = *(const float4*)(qb + (size_t)d * NPIX + mb + dq8 * 4);
      qs[myh][dq8 * 4 + 0][d] = vv.x;
      qs[myh][dq8 * 4 + 1][d] = vv.y;
      qs[myh][dq8 * 4 + 2][d] = vv.z;
      qs[myh][dq8 * 4 + 3][d] = vv.w;
    }
    __syncthreads();
    // energy: two 16x16 tiles (m-sub 0,1), K=64 in steps of 4
    v8f e0 = {}; v8f e1 = {};
#pragma unroll
    for (int kk = 0; kk < 16; ++kk) {
      int doff = kk * 4 + 2 * half;
      v2f b0 = *(const v2f*)&qs[mh][li][doff];        // ds_load_b64
      v2f b1 = *(const v2f*)&qs[mh][16 + li][doff];
      e0 = wmma_f32(a[kk], b0, e0);
      e1 = wmma_f32(a[kk], b1, e1);
    }
    // online (max,sum) per lane element
#pragma unroll
    for (int v = 0; v < 8; ++v) {
      float e = e0[v];
      float nm = fmaxf(mx[v], e);
      sm[v] = sm[v] * __expf(mx[v] - nm) + __expf(e - nm);
      mx[v] = nm;
      e = e1[v];
      nm = fmaxf(mx[v], e);
      sm[v] = sm[v] * __expf(mx[v] - nm) + __expf(e - nm);
      mx[v] = nm;
    }
  }
  // reduce the 16 column-lanes of each row (stay inside 16-lane half-groups)
#pragma unroll
  for (int v = 0; v < 8; ++v) {
#pragma unroll
    for (int off = 8; off > 0; off >>= 1) {
      float omx = __shfl_xor(mx[v], off, 16);
      float osm = __shfl_xor(sm[v], off, 16);
      float nm = fmaxf(mx[v], omx);
      sm[v] = sm[v] * __expf(mx[v] - nm) + osm * __expf(omx - nm);
      mx[v] = nm;
    }
  }
  if (li == 0) {
#pragma unroll
    for (int v = 0; v < 8; ++v) {
      int r = nsub * 16 + v + 8 * half;
      pmx[mh][r] = mx[v];
      psm[mh][r] = sm[v];
    }
  }
  __syncthreads();
  // merge the two m-halves, emit max and 1/sum per row
  if (t < 64) {
    float m0 = pmx[0][t], m1 = pmx[1][t];
    float s0 = psm[0][t], s1 = psm[1][t];
    float M = fmaxf(m0, m1);
    float S = s0 * __expf(m0 - M) + s1 * __expf(m1 - M);
    int n = nbT + t;
    rmax[b * NPIX + n] = M;
    rinv[b * NPIX + n] = 1.0f / S;
  }
}

// ---------------------------------------------------------------------------
// Kernel 3: fused attention-apply.  grid.z selects {out, context_flow}.
// 256 threads = 8 waves.  Each wave: energy tile (nsub=w&3, msub=w>>2) -> P
// via LDS; PV accumulation over 2 channel-subtiles (c = 32w..32w+31) x 4 n-subtiles.
// ---------------------------------------------------------------------------
__global__ void attnpv_kernel(const float* __restrict__ q, const float* __restrict__ x,
                              const float* __restrict__ pre, const float* __restrict__ mask,
                              const float* __restrict__ rmax, const float* __restrict__ rinv,
                              const float* __restrict__ gamma_p, const float* __restrict__ alpha_p,
                              float* __restrict__ out) {
  __shared__ float qs[32][68];       // Q m-tile [m][d]
  __shared__ float vls[CCH][36];     // V m-tile transposed [c][m]
  __shared__ float pls[4][16][34];   // normalized P tiles [nsub][n][m]
  const int b   = blockIdx.y;
  const int nbT = blockIdx.x * 64;
  const int grp = blockIdx.z;
  const int t   = threadIdx.x;
  const int w = t >> 5, l = t & 31;
  const int half = l >> 4, li = l & 15;
  const int nsub = w & 3, msub = w >> 2;
  const float* qb = q + (size_t)b * DQ * NPIX;
  const float* V  = (grp == 0 ? x : pre) + (size_t)b * CCH * NPIX;

  // A operand: this wave's 16 query rows (all d), resident in regs
  v2f aq[16];
  {
    int nrow = nbT + nsub * 16 + li;
#pragma unroll
    for (int kk = 0; kk < 16; ++kk) {
      int d0 = kk * 4 + 2 * half;
      aq[kk].x = qb[(size_t)d0 * NPIX + nrow];
      aq[kk].y = qb[(size_t)(d0 + 1) * NPIX + nrow];
    }
  }
  // softmax stats for the 8 rows this lane owns in the energy D-tile
  float mx_r[8], inv_r[8];
#pragma unroll
  for (int v = 0; v < 8; ++v) {
    int n = nbT + nsub * 16 + v + 8 * half;
    mx_r[v]  = rmax[b * NPIX + n];
    inv_r[v] = rinv[b * NPIX + n];
  }
  v8f acc[2][4];
#pragma unroll
  for (int cs = 0; cs < 2; ++cs)
#pragma unroll
    for (int ns = 0; ns < 4; ++ns) acc[cs][ns] = (v8f){};

  const int cwb = w * 32;          // this wave's channel base
  const int dq8 = t & 7;           // staging: m-quad

  for (int it = 0; it < 128; ++it) {
    const int mb = it * 32;
    __syncthreads();
    // stage Q m-tile [m][d]
#pragma unroll
    for (int i = 0; i < 2; ++i) {
      int d = (t >> 3) + 32 * i;
      float4 vv = *(const float4*)(qb + (size_t)d * NPIX + mb + dq8 * 4);
      qs[dq8 * 4 + 0][d] = vv.x;
      qs[dq8 * 4 + 1][d] = vv.y;
      qs[dq8 * 4 + 2][d] = vv.z;
      qs[dq8 * 4 + 3][d] = vv.w;
    }
    // stage V m-tile [c][m] (rows contiguous in global -> coalesced b128)
#pragma unroll
    for (int i = 0; i < 8; ++i) {
      int c = (t >> 3) + 32 * i;
      float4 vv = *(const float4*)(V + (size_t)c * NPIX + mb + dq8 * 4);
      *(float4*)&vls[c][dq8 * 4] = vv;  // 16B-aligned (pitch 36 floats = 144B)
    }
    __syncthreads();
    // energy tile for (nsub, msub)
    v8f e = {};
#pragma unroll
    for (int kk = 0; kk < 16; ++kk) {
      int doff = kk * 4 + 2 * half;
      v2f bq2 = *(const v2f*)&qs[msub * 16 + li][doff];  // ds_load_b64
      e = wmma_f32(aq[kk], bq2, e);
    }
    // P = exp(e - rowmax) * (1/rowsum) -> LDS
#pragma unroll
    for (int v = 0; v < 8; ++v) {
      float p = __expf(e[v] - mx_r[v]) * inv_r[v];
      pls[nsub][v + 8 * half][msub * 16 + li] = p;
    }
    __syncthreads();
    // PV: out[c,n] += sum_m V[c,m] * P[n,m]
#pragma unroll
    for (int kk2 = 0; kk2 < 8; ++kk2) {
      int moff = kk2 * 4 + 2 * half;
      v2f bp[4];
#pragma unroll
      for (int ns = 0; ns < 4; ++ns)
        bp[ns] = *(const v2f*)&pls[ns][li][moff];        // B = P^T, ds_load_b64
#pragma unroll
      for (int cs = 0; cs < 2; ++cs) {
        v2f av = *(const v2f*)&vls[cwb + cs * 16 + li][moff];  // A = V, ds_load_b64
#pragma unroll
        for (int ns = 0; ns < 4; ++ns)
          acc[cs][ns] = wmma_f32(av, bp[ns], acc[cs][ns]);
      }
    }
  }
  // epilogue: blend + store (coalesced over n)
  const float g  = gamma_p[0];
  const float al = alpha_p[0];
#pragma unroll
  for (int cs = 0; cs < 2; ++cs) {
#pragma unroll
    for (int ns = 0; ns < 4; ++ns) {
      int n = nbT + ns * 16 + li;
      float mval = mask[b * NPIX + n];
#pragma unroll
      for (int v = 0; v < 8; ++v) {
        int c = cwb + cs * 16 + v + 8 * half;
        float a = acc[cs][ns][v];
        float vref = V[(size_t)c * NPIX + n];
        size_t oi = ((size_t)(b * 2 * CCH + grp * CCH + c)) * NPIX + n;
        out[oi] = (grp == 0) ? fmaf(g, a, vref)
                             : al * (1.0f - mval) * a + mval * vref;
      }
    }
  }
}

extern "C" void kernel_launch(void* const* d_in, const int* in_sizes, int n_in,
                              void* d_out, int out_size, void* d_ws, size_t ws_size,
                              hipStream_t stream) {
  const float* x     = (const float*)d_in[0];
  const float* pre   = (const float*)d_in[1];
  const float* mask  = (const float*)d_in[2];
  const float* wq    = (const float*)d_in[3];
  const float* bq    = (const float*)d_in[4];
  const float* gamma = (const float*)d_in[5];
  const float* alpha = (const float*)d_in[6];
  float* out = (float*)d_out;

  float* q    = (float*)d_ws;                    // B*D*N   = 1,048,576 f
  float* rmax = q + (size_t)BATCH * DQ * NPIX;   // B*N     = 16,384 f
  float* rinv = rmax + (size_t)BATCH * NPIX;     // B*N     = 16,384 f

  qproj_kernel <<<dim3(NPIX / 16, BATCH),    256, 0, stream>>>(x, wq, bq, q);
  stats_kernel <<<dim3(NPIX / 64, BATCH),    256, 0, stream>>>(q, rmax, rinv);
  attnpv_kernel<<<dim3(NPIX / 64, BATCH, 2), 256, 0, stream>>>(q, x, pre, mask, rmax, rinv,
                                                               gamma, alpha, out);
}